// pathLoss_47090021433712
// MI455X (gfx1250) — compile-verified
//
#include <hip/hip_runtime.h>
#include <hip/hip_bf16.h>

typedef __attribute__((ext_vector_type(2))) float v2f;
typedef __attribute__((ext_vector_type(8))) float v8f;

#define B_DIM 1024
#define T_DIM 2048
#define C_DIM 5
#define NWAVE 8
#define PADT 272   // 256 elements + 16 pad (one per 16) -> stride-17 columns in LDS

__device__ __forceinline__ v8f wmma_acc(v2f a, v2f b, v8f c) {
  // D = A(16x4,f32) x B(4x16,f32) + C(16x16,f32)
  return __builtin_amdgcn_wmma_f32_16x16x4_f32(false, a, false, b, (short)0, c,
                                               false, false);
}

// Broadcast the value of `x` held by (uniform, compile-time) lane `srclane`
// to all lanes via v_readlane (no LDS pipe, no DScnt).
__device__ __forceinline__ float lane_bcast(float x, int srclane) {
  const int bits = __builtin_amdgcn_readlane(__builtin_bit_cast(int, x), srclane);
  return __builtin_bit_cast(float, bits);
}

// Inclusive 256-wide prefix sum of a wave tile stored in LDS.
// tile[i + (i>>4)] holds v[i], i = m + 16*n (column-major 16x16).
// Returns P in the WMMA C-layout: lane l, slot r  <->  element (m = r + 8*(l>>4), n = l&15).
__device__ __forceinline__ v8f wave_scan256(const float* tile, int h, int n, int ma) {
  v8f R = {};
  // Stage 1: R = L * M  (column-wise inclusive prefix), K=16 chained as 4x K=4.
  #pragma unroll
  for (int k = 0; k < 4; ++k) {
    const int q0 = 4 * k + 2 * h;             // K-index held by VGPR0 for this lane-half
    v2f a, b;
    a.x = (q0     <= ma) ? 1.0f : 0.0f;       // L[m][q0]
    a.y = (q0 + 1 <= ma) ? 1.0f : 0.0f;       // L[m][q0+1]
    b.x = tile[q0     + 17 * n];              // M[q0][n]   = v[q0   + 16n]
    b.y = tile[q0 + 1 + 17 * n];              // M[q0+1][n]
    R = wmma_acc(a, b, R);
  }
  // Column sums live in row m=15: slot r=7, lanes 16..31 (lane 16+q holds colsum[q]).
  float cs[16];
  #pragma unroll
  for (int q = 0; q < 16; ++q) cs[q] = lane_bcast(R[7], 16 + q);
  // Stage 2: P = A2 * B2 + R, (A2*B2)[m][n] = sum_{q<n} colsum[q]  (carry across columns)
  v8f P = R;
  #pragma unroll
  for (int k = 0; k < 4; ++k) {
    const int q0 = 4 * k + 2 * h;
    v2f a, b;
    a.x = h ? cs[4 * k + 2] : cs[4 * k + 0];  // colsum[q0]
    a.y = h ? cs[4 * k + 3] : cs[4 * k + 1];  // colsum[q0+1]
    b.x = (q0     < n) ? 1.0f : 0.0f;         // strict upper-triangular ones
    b.y = (q0 + 1 < n) ? 1.0f : 0.0f;
    P = wmma_acc(a, b, P);
  }
  return P;
}

__global__ __launch_bounds__(256)
void path_loss_kernel(const float* __restrict__ outputs,
                      const int* __restrict__ labels,
                      float* __restrict__ partial) {
  __shared__ float s_dthp[NWAVE * PADT];
  __shared__ float s_dtht[NWAVE * PADT];
  __shared__ float s_fwp [NWAVE * PADT];
  __shared__ float s_fwt [NWAVE * PADT];
  __shared__ float s_totp[NWAVE], s_tott[NWAVE], s_totd[NWAVE];
  __shared__ float s_red [NWAVE];

  const int b    = blockIdx.x;
  const int tid  = threadIdx.x;
  const int lane = tid & 31;
  const int wv   = tid >> 5;
  const int h    = (lane >> 4) & 1;
  const int n    = lane & 15;
  const int ma   = lane & 15;

  const size_t rowO = (size_t)b * T_DIM * C_DIM;
  const size_t rowL = (size_t)b * T_DIM;
  const float  WT   = 1.0f / (1.0f + 1e-6f);   // true-path weight

  // ---- Phase 1: elementwise fw/dth for both paths (coalesced), stash in LDS ----
  #pragma unroll
  for (int j = 0; j < 8; ++j) {
    const int t = tid + 256 * j;
    const float* o = outputs + rowO + (size_t)t * C_DIM;
    if (j < 7) __builtin_prefetch(o + 256 * C_DIM, 0, 1);   // global_prefetch_b8
    const float o0 = o[0], o1 = o[1], o2 = o[2], o3 = o[3], o4 = o[4];
    int pred = 0; float mx = o0;                 // first-max argmax (strict >)
    if (o1 > mx) { mx = o1; pred = 1; }
    if (o2 > mx) { mx = o2; pred = 2; }
    if (o3 > mx) { mx = o3; pred = 3; }
    if (o4 > mx) { mx = o4; pred = 4; }
    const float se = __expf(o0 - mx) + __expf(o1 - mx) + __expf(o2 - mx) +
                     __expf(o3 - mx) + __expf(o4 - mx);
    const float s  = 1.0f / se;                  // softmax value at argmax
    const float wp = s / (s + 1e-6f);
    const float fwp  = (pred == 0 || pred == 3 || pred == 4) ? wp : 0.0f;
    const float sgnp = (pred == 1 || pred == 3) ? 1.0f
                     : ((pred == 2 || pred == 4) ? -1.0f : 0.0f);
    const float dthp = 0.15f * wp * sgnp;

    const int lb = labels[rowL + t];
    const float fwt  = (lb == 0 || lb == 3 || lb == 4) ? WT : 0.0f;
    const float sgnt = (lb == 1 || lb == 3) ? 1.0f
                     : ((lb == 2 || lb == 4) ? -1.0f : 0.0f);
    const float dtht = 0.15f * WT * sgnt;

    const int off = t + (t >> 4);
    s_dthp[off] = dthp;
    s_dtht[off] = dtht;
    s_fwp [off] = fwp;
    s_fwt [off] = fwt;
  }
  __syncthreads();

  // ---- Phase 2: per-wave WMMA scans of dtheta (both paths) ----
  float* tp = s_dthp + PADT * wv;
  float* tt = s_dtht + PADT * wv;
  const v8f Pp = wave_scan256(tp, h, n, ma);
  const v8f Pt = wave_scan256(tt, h, n, ma);
  if (lane == 31) { s_totp[wv] = Pp[7]; s_tott[wv] = Pt[7]; }  // tile totals (i=255)
  __syncthreads();

  float carp = 0.0f, cart = 0.0f;
  for (int w2 = 0; w2 < wv; ++w2) { carp += s_totp[w2]; cart += s_tott[w2]; }

  // ---- Phase 3: theta (exclusive) -> d = fw_p*cos(th_p) - fw_t*cos(th_t) ----
  const float* fp = s_fwp + PADT * wv;
  const float* ft = s_fwt + PADT * wv;
  #pragma unroll
  for (int r = 0; r < 8; ++r) {
    const int i   = r + 8 * h + 16 * n;   // element index in tile (C-layout)
    const int off = i + n;                // padded LDS offset (i>>4 == n here)
    const float thp = carp + Pp[r] - tp[off];   // theta read before update
    const float tht = cart + Pt[r] - tt[off];
    const float dv  = fp[off] * __cosf(thp) - ft[off] * __cosf(tht);
    tp[off] = dv;                         // reuse dth_p tile for the d-scan
  }
  __syncthreads();

  // ---- Phase 4: WMMA scan of d -> dx (inclusive), accumulate loss ----
  const v8f Pd = wave_scan256(tp, h, n, ma);
  if (lane == 31) s_totd[wv] = Pd[7];
  __syncthreads();
  float card = 0.0f;
  for (int w2 = 0; w2 < wv; ++w2) card += s_totd[w2];

  float local = 0.0f;
  #pragma unroll
  for (int r = 0; r < 8; ++r) {
    const float dx = card + Pd[r];
    local += sqrtf(2.0f * dx * dx + 1e-12f);
  }
  #pragma unroll
  for (int o2 = 16; o2 > 0; o2 >>= 1) local += __shfl_down(local, o2, 32);
  if (lane == 0) s_red[wv] = local;
  __syncthreads();
  if (tid == 0) {
    float sum = 1e-6f;                    // path column t=0: sqrt(1e-12)
    for (int w2 = 0; w2 < NWAVE; ++w2) sum += s_red[w2];
    partial[b] = sum;
  }
}

__global__ __launch_bounds__(256)
void path_loss_reduce(const float* __restrict__ partial, float* __restrict__ out) {
  __shared__ float s_red[8];
  const int tid = threadIdx.x;
  float v = partial[tid] + partial[tid + 256] + partial[tid + 512] + partial[tid + 768];
  #pragma unroll
  for (int o2 = 16; o2 > 0; o2 >>= 1) v += __shfl_down(v, o2, 32);
  if ((tid & 31) == 0) s_red[tid >> 5] = v;
  __syncthreads();
  if (tid == 0) {
    float s = 0.0f;
    for (int i = 0; i < 8; ++i) s += s_red[i];
    out[0] = s / (1024.0f * 2049.0f);    // mean over B x (T+1)
  }
}

extern "C" void kernel_launch(void* const* d_in, const int* in_sizes, int n_in,
                              void* d_out, int out_size, void* d_ws, size_t ws_size,
                              hipStream_t stream) {
  const float* outputs = (const float*)d_in[0];
  const int*   labels  = (const int*)d_in[1];
  (void)in_sizes; (void)n_in; (void)out_size; (void)ws_size;
  float* partial = (float*)d_ws;           // 1024 floats of scratch
  path_loss_kernel<<<B_DIM, 256, 0, stream>>>(outputs, labels, partial);
  path_loss_reduce<<<1, 256, 0, stream>>>(partial, (float*)d_out);
}